// MPNN_23313082483685
// MI455X (gfx1250) — compile-verified
//
#include <hip/hip_runtime.h>
#include <math.h>

typedef float v2f __attribute__((ext_vector_type(2)));
typedef float v8f __attribute__((ext_vector_type(8)));

#define NWAVE 8
#define NORB  32
#define NANG  9
#define CCD   25          // 3*NWAVE+1
#define COD   72          // NANG*NWAVE

__device__ __forceinline__ v8f wmma_f32_16x16x4(v2f a, v2f b, v8f c) {
  // D = A(16x4 f32) * B(4x16 f32) + C(16x16 f32), wave32
  return __builtin_amdgcn_wmma_f32_16x16x4_f32(false, a, false, b, (short)0, c,
                                               false, false);
}

__device__ __forceinline__ float silu_f(float x) { return x / (1.0f + expf(-x)); }

// ---------------- K1: per-atom embedding MLP (1 -> 8 -> 8 -> 25) ----------------
__global__ void embed_kernel(const float* __restrict__ species,
                             const float* __restrict__ W1, const float* __restrict__ b1,
                             const float* __restrict__ W2, const float* __restrict__ b2,
                             const float* __restrict__ Wo, const float* __restrict__ bo,
                             float* __restrict__ cc, float* __restrict__ icoef,
                             float* __restrict__ density, int N)
{
  int n = blockIdx.x * blockDim.x + threadIdx.x;
  if (n >= N) return;
  float x = species[n];
  float h[8], h2[8];
  #pragma unroll
  for (int j = 0; j < 8; ++j) h[j] = silu_f(x * W1[j] + b1[j]);
  {
    float mu = 0.f;
    #pragma unroll
    for (int j = 0; j < 8; ++j) mu += h[j];
    mu *= 0.125f;
    float var = 0.f;
    #pragma unroll
    for (int j = 0; j < 8; ++j) { float d = h[j] - mu; var += d * d; }
    var *= 0.125f;
    float ri = rsqrtf(var + 1e-5f);
    #pragma unroll
    for (int j = 0; j < 8; ++j) h[j] = (h[j] - mu) * ri;
  }
  #pragma unroll
  for (int j = 0; j < 8; ++j) {
    float s = b2[j];
    #pragma unroll
    for (int i = 0; i < 8; ++i) s += h[i] * W2[i * 8 + j];
    h2[j] = silu_f(s);
  }
  {
    float mu = 0.f;
    #pragma unroll
    for (int j = 0; j < 8; ++j) mu += h2[j];
    mu *= 0.125f;
    float var = 0.f;
    #pragma unroll
    for (int j = 0; j < 8; ++j) { float d = h2[j] - mu; var += d * d; }
    var *= 0.125f;
    float ri = rsqrtf(var + 1e-5f);
    #pragma unroll
    for (int j = 0; j < 8; ++j) h2[j] = (h2[j] - mu) * ri;
  }
  #pragma unroll
  for (int j = 0; j < CCD; ++j) {
    float s = bo[j];
    #pragma unroll
    for (int i = 0; i < 8; ++i) s += h2[i] * Wo[i * CCD + j];
    cc[(size_t)n * CCD + j] = s;
    if (j >= 2 * NWAVE && j < 3 * NWAVE) icoef[(size_t)n * NWAVE + (j - 2 * NWAVE)] = s;
  }
  #pragma unroll
  for (int m = 0; m < NORB; ++m) density[(size_t)n * NORB + m] = 0.f;
}

// ---------------- K2: per-edge geometry (cut, sph, radial; layer-invariant) ----------------
__global__ __launch_bounds__(256)
void edge_pre_kernel(const float* __restrict__ cart, const int* __restrict__ nl,
                     const float* __restrict__ shifts, const float* __restrict__ nf,
                     const float* __restrict__ cc,
                     float* __restrict__ cut, float* __restrict__ radial,
                     float* __restrict__ sph, int NE)
{
  int e = blockIdx.x * blockDim.x + threadIdx.x;
  if (e >= NE) return;
  int ic = nl[e], in_ = nl[NE + e];
  float x = cart[3 * in_ + 0] - cart[3 * ic + 0] + shifts[3 * e + 0];
  float y = cart[3 * in_ + 1] - cart[3 * ic + 1] + shifts[3 * e + 1];
  float z = cart[3 * in_ + 2] - cart[3 * ic + 2] + shifts[3 * e + 2];
  float r2 = x * x + y * y + z * z;
  float d = sqrtf(r2);
  float cf = 0.5f * cosf(d * 0.78539816339744830961f) + 0.5f;  // pi/CUTOFF = pi/4
  float cv = nf[e] * cf * cf;
  cut[e] = cv;
  const float c0 = 0.28209479177387814f, c1 = 0.4886025119029199f;
  const float c2a = 1.0925484305920792f, c2b = 0.31539156525252005f;
  const float c2c = 0.5462742152960396f;
  sph[(size_t)0 * NE + e] = c0;
  sph[(size_t)1 * NE + e] = c1 * y;
  sph[(size_t)2 * NE + e] = c1 * z;
  sph[(size_t)3 * NE + e] = c1 * x;
  sph[(size_t)4 * NE + e] = c2a * x * y;
  sph[(size_t)5 * NE + e] = c2a * y * z;
  sph[(size_t)6 * NE + e] = c2b * (3.0f * z * z - r2);
  sph[(size_t)7 * NE + e] = c2a * x * z;
  sph[(size_t)8 * NE + e] = c2c * (x * x - y * y);
  const float* ccr = cc + (size_t)in_ * CCD;
  #pragma unroll
  for (int w = 0; w < NWAVE; ++w) {
    float alpha = ccr[w], rs = ccr[NWAVE + w];
    float dd = d - rs;
    radial[(size_t)w * NE + e] = cv * expf(-alpha * dd * dd);
  }
}

// ---------------- K3: per-edge message + scatter-add into orbitals ----------------
__global__ __launch_bounds__(256)
void edge_message_kernel(const int* __restrict__ nl, int NE,
                         const float* __restrict__ cut,
                         const float* __restrict__ radial,
                         const float* __restrict__ sph,
                         const float* __restrict__ icoef,
                         const float* __restrict__ co_old,
                         float* __restrict__ co_new)
{
  int e = blockIdx.x * blockDim.x + threadIdx.x;
  if (e >= NE) return;
  int ic = nl[e], in_ = nl[NE + e];
  float c = cut[e];
  const float4 i0 = *(const float4*)(icoef + (size_t)in_ * NWAVE);
  const float4 i1 = *(const float4*)(icoef + (size_t)in_ * NWAVE + 4);
  float rp[NWAVE];
  rp[0] = radial[(size_t)0 * NE + e] * i0.x;
  rp[1] = radial[(size_t)1 * NE + e] * i0.y;
  rp[2] = radial[(size_t)2 * NE + e] * i0.z;
  rp[3] = radial[(size_t)3 * NE + e] * i0.w;
  rp[4] = radial[(size_t)4 * NE + e] * i1.x;
  rp[5] = radial[(size_t)5 * NE + e] * i1.y;
  rp[6] = radial[(size_t)6 * NE + e] * i1.z;
  rp[7] = radial[(size_t)7 * NE + e] * i1.w;
  const float* cr = co_old + (size_t)in_ * COD;
  float* cw = co_new + (size_t)ic * COD;
  #pragma unroll
  for (int k = 0; k < NANG; ++k) {
    float s = sph[(size_t)k * NE + e];
    float4 a = *(const float4*)(cr + k * NWAVE);
    float4 b = *(const float4*)(cr + k * NWAVE + 4);
    unsafeAtomicAdd(cw + k * NWAVE + 0, rp[0] * s + a.x * c);
    unsafeAtomicAdd(cw + k * NWAVE + 1, rp[1] * s + a.y * c);
    unsafeAtomicAdd(cw + k * NWAVE + 2, rp[2] * s + a.z * c);
    unsafeAtomicAdd(cw + k * NWAVE + 3, rp[3] * s + a.w * c);
    unsafeAtomicAdd(cw + k * NWAVE + 4, rp[4] * s + b.x * c);
    unsafeAtomicAdd(cw + k * NWAVE + 5, rp[5] * s + b.y * c);
    unsafeAtomicAdd(cw + k * NWAVE + 6, rp[6] * s + b.z * c);
    unsafeAtomicAdd(cw + k * NWAVE + 7, rp[7] * s + b.w * c);
  }
}

// ---------------- K4: per-atom contraction + WMMA MLP ----------------
// 4 waves/block, each wave owns 16 atoms. fp32 WMMA 16x16x4 for all GEMMs.
__global__ __launch_bounds__(128)
void atom_update_kernel(const float* __restrict__ co,     // [N,72] updated orbitals
                        const float* __restrict__ ccoef,  // [8,32]
                        const float* __restrict__ cc,     // [N,25]
                        float* __restrict__ density,      // [N,32] in/out
                        const float* __restrict__ W1, const float* __restrict__ B1,
                        const float* __restrict__ W2, const float* __restrict__ B2,
                        const float* __restrict__ Wo, const float* __restrict__ Bo,
                        int nout,
                        float* __restrict__ iter_out,     // [N,8] when nout==8
                        const float* __restrict__ center_factor,
                        float* __restrict__ out_scalar,   // when nout==1
                        int N)
{
  __shared__ float X[64][36];    // density tile  (stride 36: conflict-free col reads)
  __shared__ float H1[64][68];   // hidden1       (stride 68)
  __shared__ float H2[64][68];   // hidden2
  __shared__ float red[128];

  const int tid  = threadIdx.x;
  const int wv   = tid >> 5;
  const int lane = tid & 31;
  const int half = lane >> 4;        // WMMA lane-half (K / row split)
  const int l    = lane & 15;
  const int wrow = wv * 16;
  const int abase = blockIdx.x * 64 + wrow;

  // --- Stage A: contracted = co @ ccoef ; density += sum_k contracted^2 * cc[:,24]
  {
    const int m = lane;  // orbital column 0..31
    float col[NWAVE];
    #pragma unroll
    for (int w = 0; w < NWAVE; ++w) col[w] = ccoef[w * NORB + m];
    for (int a = 0; a < 16; ++a) {
      int n  = abase + a;
      int nc = (n < N) ? n : (N - 1);
      const float* crow = co + (size_t)nc * COD;
      float t = 0.f;
      #pragma unroll
      for (int k = 0; k < NANG; ++k) {
        float s = 0.f;
        #pragma unroll
        for (int w = 0; w < NWAVE; ++w) s += crow[k * NWAVE + w] * col[w];
        t += s * s;
      }
      float dnew = density[(size_t)nc * NORB + m] + t * cc[(size_t)nc * CCD + 24];
      if (n < N) density[(size_t)n * NORB + m] = dnew;
      X[wrow + a][m] = dnew;
    }
  }
  __syncthreads();

  // --- GEMM1: X[16x32] @ W1[32x64] + b1, SiLU -> H1
  #pragma unroll
  for (int tN = 0; tN < 4; ++tN) {
    v8f acc = {0.f, 0.f, 0.f, 0.f, 0.f, 0.f, 0.f, 0.f};
    #pragma unroll
    for (int k0 = 0; k0 < NORB; k0 += 4) {
      v2f av, bv;
      av.x = X[wrow + l][k0 + 2 * half + 0];
      av.y = X[wrow + l][k0 + 2 * half + 1];
      bv.x = W1[(size_t)(k0 + 2 * half + 0) * 64 + tN * 16 + l];
      bv.y = W1[(size_t)(k0 + 2 * half + 1) * 64 + tN * 16 + l];
      acc = wmma_f32_16x16x4(av, bv, acc);
    }
    float bias = B1[tN * 16 + l];
    #pragma unroll
    for (int r = 0; r < 8; ++r)
      H1[wrow + r + 8 * half][tN * 16 + l] = silu_f(acc[r] + bias);
  }
  __syncthreads();

  // --- LayerNorm rows of H1
  if (half == 0) {
    float* row = H1[wrow + l];
    float mu = 0.f;
    for (int c2 = 0; c2 < 64; ++c2) mu += row[c2];
    mu *= (1.f / 64.f);
    float var = 0.f;
    for (int c2 = 0; c2 < 64; ++c2) { float dd = row[c2] - mu; var += dd * dd; }
    var *= (1.f / 64.f);
    float ri = rsqrtf(var + 1e-5f);
    for (int c2 = 0; c2 < 64; ++c2) row[c2] = (row[c2] - mu) * ri;
  }
  __syncthreads();

  // --- GEMM2: H1[16x64] @ W2[64x64] + b2, SiLU -> H2
  #pragma unroll
  for (int tN = 0; tN < 4; ++tN) {
    v8f acc = {0.f, 0.f, 0.f, 0.f, 0.f, 0.f, 0.f, 0.f};
    #pragma unroll
    for (int k0 = 0; k0 < 64; k0 += 4) {
      v2f av, bv;
      av.x = H1[wrow + l][k0 + 2 * half + 0];
      av.y = H1[wrow + l][k0 + 2 * half + 1];
      bv.x = W2[(size_t)(k0 + 2 * half + 0) * 64 + tN * 16 + l];
      bv.y = W2[(size_t)(k0 + 2 * half + 1) * 64 + tN * 16 + l];
      acc = wmma_f32_16x16x4(av, bv, acc);
    }
    float bias = B2[tN * 16 + l];
    #pragma unroll
    for (int r = 0; r < 8; ++r)
      H2[wrow + r + 8 * half][tN * 16 + l] = silu_f(acc[r] + bias);
  }
  __syncthreads();

  // --- LayerNorm rows of H2
  if (half == 0) {
    float* row = H2[wrow + l];
    float mu = 0.f;
    for (int c2 = 0; c2 < 64; ++c2) mu += row[c2];
    mu *= (1.f / 64.f);
    float var = 0.f;
    for (int c2 = 0; c2 < 64; ++c2) { float dd = row[c2] - mu; var += dd * dd; }
    var *= (1.f / 64.f);
    float ri = rsqrtf(var + 1e-5f);
    for (int c2 = 0; c2 < 64; ++c2) row[c2] = (row[c2] - mu) * ri;
  }
  __syncthreads();

  // --- Output layer
  if (nout == NWAVE) {
    // H2[16x64] @ Wo[64x8] + bo -> iter_coeff (WMMA tile, cols 8..15 zero-padded)
    v8f acc = {0.f, 0.f, 0.f, 0.f, 0.f, 0.f, 0.f, 0.f};
    #pragma unroll
    for (int k0 = 0; k0 < 64; k0 += 4) {
      v2f av, bv;
      av.x = H2[wrow + l][k0 + 2 * half + 0];
      av.y = H2[wrow + l][k0 + 2 * half + 1];
      bv.x = (l < NWAVE) ? Wo[(size_t)(k0 + 2 * half + 0) * NWAVE + l] : 0.f;
      bv.y = (l < NWAVE) ? Wo[(size_t)(k0 + 2 * half + 1) * NWAVE + l] : 0.f;
      acc = wmma_f32_16x16x4(av, bv, acc);
    }
    if (l < NWAVE) {
      float bias = Bo[l];
      #pragma unroll
      for (int r = 0; r < 8; ++r) {
        int row = wrow + r + 8 * half;
        int n = blockIdx.x * 64 + row;
        if (n < N) iter_out[(size_t)n * NWAVE + l] = acc[r] + bias;
      }
    }
  } else {
    // final: out0 = H2 @ Wo[64x1] + bo; reduce sum(out0 * center_factor)
    float partial = 0.f;
    if (half == 0) {
      int n = abase + l;
      if (n < N) {
        const float* row = H2[wrow + l];
        float s = 0.f;
        for (int c2 = 0; c2 < 64; ++c2) s += row[c2] * Wo[c2];
        s += Bo[0];
        partial = s * center_factor[n];
      }
    }
    red[tid] = partial;
    __syncthreads();
    if (tid == 0) {
      float s = 0.f;
      for (int i2 = 0; i2 < 128; ++i2) s += red[i2];
      unsafeAtomicAdd(out_scalar, s);
    }
  }
}

extern "C" void kernel_launch(void* const* d_in, const int* in_sizes, int n_in,
                              void* d_out, int out_size, void* d_ws, size_t ws_size,
                              hipStream_t stream) {
  const float* cart          = (const float*)d_in[0];
  const int*   nl            = (const int*)d_in[1];
  const float* shifts        = (const float*)d_in[2];
  const float* center_factor = (const float*)d_in[3];
  const float* neigh_factor  = (const float*)d_in[4];
  const float* species       = (const float*)d_in[5];
  const float* ccoef         = (const float*)d_in[6];
  const float* eW1 = (const float*)d_in[7];
  const float* eb1 = (const float*)d_in[8];
  const float* eW2 = (const float*)d_in[9];
  const float* eb2 = (const float*)d_in[10];
  const float* eWo = (const float*)d_in[11];
  const float* ebo = (const float*)d_in[12];
  const float* mW1 = (const float*)d_in[13];
  const float* mb1 = (const float*)d_in[14];
  const float* mW2 = (const float*)d_in[15];
  const float* mb2 = (const float*)d_in[16];
  const float* mWo = (const float*)d_in[17];
  const float* mbo = (const float*)d_in[18];

  const int N  = in_sizes[3];   // N_ATOMS (center_factor)
  const int NE = in_sizes[4];   // N_EDGES (neigh_factor)

  float* ws = (float*)d_ws;
  size_t off = 0;
  auto alloc = [&](size_t cnt) {
    float* p = ws + off;
    off += (cnt + 63) & ~(size_t)63;   // 256B-aligned regions
    return p;
  };
  float* cc      = alloc((size_t)N * CCD);
  float* icoef   = alloc((size_t)N * NWAVE);
  float* density = alloc((size_t)N * NORB);
  float* coA     = alloc((size_t)N * COD);
  float* coB     = alloc((size_t)N * COD);
  float* cut     = alloc((size_t)NE);
  float* radial  = alloc((size_t)NE * NWAVE);
  float* sph     = alloc((size_t)NE * NANG);
  (void)ws_size; (void)n_in; (void)out_size;

  hipMemsetAsync(d_out, 0, sizeof(float), stream);
  hipMemsetAsync(coA, 0, (size_t)N * COD * sizeof(float), stream);

  embed_kernel<<<(N + 127) / 128, 128, 0, stream>>>(
      species, eW1, eb1, eW2, eb2, eWo, ebo, cc, icoef, density, N);

  edge_pre_kernel<<<(NE + 255) / 256, 256, 0, stream>>>(
      cart, nl, shifts, neigh_factor, cc, cut, radial, sph, NE);

  float* co_old = coA;
  float* co_new = coB;
  const size_t coBytes = (size_t)N * COD * sizeof(float);
  const int gridE = (NE + 255) / 256;
  const int gridA = (N + 63) / 64;

  for (int li = 0; li < 4; ++li) {
    hipMemcpyAsync(co_new, co_old, coBytes, hipMemcpyDeviceToDevice, stream);
    edge_message_kernel<<<gridE, 256, 0, stream>>>(
        nl, NE, cut, radial, sph, icoef, co_old, co_new);

    const float *W1, *B1, *W2, *B2, *Wo, *Bo;
    int nout;
    if (li < 3) {
      W1 = mW1 + (size_t)li * NORB * 64;  B1 = mb1 + (size_t)li * 64;
      W2 = mW2 + (size_t)li * 64 * 64;    B2 = mb2 + (size_t)li * 64;
      Wo = mWo + (size_t)li * 64 * NWAVE; Bo = mbo + (size_t)li * NWAVE;
      nout = NWAVE;
    } else {
      W1 = (const float*)d_in[19]; B1 = (const float*)d_in[20];
      W2 = (const float*)d_in[21]; B2 = (const float*)d_in[22];
      Wo = (const float*)d_in[23]; Bo = (const float*)d_in[24];
      nout = 1;
    }
    atom_update_kernel<<<gridA, 128, 0, stream>>>(
        co_new, ccoef, cc, density, W1, B1, W2, B2, Wo, Bo,
        nout, icoef, center_factor, (float*)d_out, N);

    float* t = co_old; co_old = co_new; co_new = t;
  }
}